// QNN_79139067396105
// MI455X (gfx1250) — compile-verified
//
#include <hip/hip_runtime.h>
#include <hip/hip_bf16.h>

typedef __attribute__((ext_vector_type(2))) float v2f;
typedef __attribute__((ext_vector_type(8))) float v8f;

#define DEPTH 15
#define QUBITS 12

// ---------------------------------------------------------------------------
// Prep kernel: build fused 16x16 complex unitaries per (layer, group of 4
// wires).  ws layout per (layer,group): [Ure(256) | Uim(256) | negUim(256)].
// U[r,c] = prod_{q=0..3} R_{wire=4g+q}[bit_{3-q}(r), bit_{3-q}(c)]
// Rot(phi,theta,omega) = [[e^{-i(phi+om)/2} c, -e^{i(phi-om)/2} s],
//                         [e^{-i(phi-om)/2} s,  e^{i(phi+om)/2} c]]
// ---------------------------------------------------------------------------
__global__ void qnn_prep(const float* __restrict__ w, float* __restrict__ ws) {
    int layer = blockIdx.x / 3;
    int g     = blockIdx.x % 3;
    int e     = threadIdx.x;          // 0..255 : one matrix entry
    int r     = e >> 4;
    int cc    = e & 15;

    float pre = 1.f, pim = 0.f;
    for (int q = 0; q < 4; ++q) {
        int y = 4 * g + q;
        const float* wy = w + (layer * QUBITS + y) * 3;
        float phi = wy[0], th = wy[1], om = wy[2];
        float c = __cosf(th * 0.5f), s = __sinf(th * 0.5f);
        int br = (r >> (3 - q)) & 1;
        int bc = (cc >> (3 - q)) & 1;
        float ang, mag;
        if (br == 0 && bc == 0)      { ang = -0.5f * (phi + om); mag =  c; }
        else if (br == 0 && bc == 1) { ang =  0.5f * (phi - om); mag = -s; }
        else if (br == 1 && bc == 0) { ang = -0.5f * (phi - om); mag =  s; }
        else                         { ang =  0.5f * (phi + om); mag =  c; }
        float ere = mag * __cosf(ang);
        float eim = mag * __sinf(ang);
        float nre = pre * ere - pim * eim;
        float nim = pre * eim + pim * ere;
        pre = nre; pim = nim;
    }
    float* base = ws + (size_t)(layer * 3 + g) * 768;
    base[e]       = pre;    // U real
    base[256 + e] = pim;    // U imag
    base[512 + e] = -pim;   // -U imag
}

// ---------------------------------------------------------------------------
// Main kernel: one workgroup (256 threads = 8 wave32) per batch element.
// State vector (4096 complex) lives in LDS as split re/im, double-buffered.
// Each layer: 3 group-unitaries via V_WMMA_F32_16X16X4_F32 (16 WMMAs per
// complex 16x16x16 tile); the CNOT-ring permutation is fused into the last
// group's writeback through PI[].  Each wave owns 2 of the 16 column tiles.
// ---------------------------------------------------------------------------
__global__ void __launch_bounds__(256) qnn_main(const float* __restrict__ x,
                                                const float* __restrict__ ws,
                                                float* __restrict__ out) {
    // S: [buf0 re | buf0 im | buf1 re | buf1 im], each 4096 floats
    __shared__ float S[16384];
    __shared__ unsigned short PI[4096];
    __shared__ float rbuf[260];

    const int b    = blockIdx.x;
    const int tid  = threadIdx.x;
    const int lane = tid & 31;
    const int wave = tid >> 5;       // 0..7

    // ---- inverse CNOT-ring permutation table (data at j moves to PI[j]) ----
    for (int i = tid; i < 4096; i += 256) {
        int j = i;
        #pragma unroll
        for (int y = 0; y < 12; ++y) {
            int t  = (y + 1) == 12 ? 0 : (y + 1);
            int cb = (j >> (11 - y)) & 1;
            j ^= cb << (11 - t);
        }
        PI[i] = (unsigned short)j;
    }

    // ---- load input, compute L2 norm ----
    const float* xb = x + (size_t)b * 1024;
    float xv[4];
    float loc = 0.f;
    #pragma unroll
    for (int k = 0; k < 4; ++k) {
        xv[k] = xb[tid + 256 * k];
        loc += xv[k] * xv[k];
    }
    rbuf[tid] = loc;
    __syncthreads();
    for (int off = 128; off > 0; off >>= 1) {
        if (tid < off) rbuf[tid] += rbuf[tid + off];
        __syncthreads();
    }
    float ssq = rbuf[0];
    float inv = rsqrtf(ssq > 0.f ? ssq : 1.f);
    __syncthreads();

    // ---- initialize state in buffer 0: amp[4d + 0] = x[d]/||x||, rest 0 ----
    for (int i = tid; i < 4096; i += 256) { S[i] = 0.f; S[4096 + i] = 0.f; }
    __syncthreads();
    #pragma unroll
    for (int k = 0; k < 4; ++k) {
        int d = tid + 256 * k;
        S[4 * d] = xv[k] * inv;
    }
    __syncthreads();

    const int r    = lane & 15;           // fragment row / column within tile
    const int colb = (lane >> 4) << 1;    // K sub-offset: 0 (lanes<16) or 2
    const int mb   = (lane >> 4) << 3;    // D row base:  0 or 8

    for (int layer = 0; layer < DEPTH; ++layer) {
        const int cur = (layer & 1) * 8192;
        const int nxt = 8192 - cur;

        for (int g = 0; g < 3; ++g) {
            // ---- load A fragments (U, U_im, -U_im), 16x16x4 layout ----
            const float* U = ws + (size_t)(layer * 3 + g) * 768;
            v2f are[4], aim[4], anm[4];
            #pragma unroll
            for (int kk = 0; kk < 4; ++kk) {
                int cb = 4 * kk + colb;
                v2f t0, t1, t2;
                t0[0] = U[r * 16 + cb];       t0[1] = U[r * 16 + cb + 1];
                t1[0] = U[256 + r * 16 + cb]; t1[1] = U[256 + r * 16 + cb + 1];
                t2[0] = U[512 + r * 16 + cb]; t2[1] = U[512 + r * 16 + cb + 1];
                are[kk] = t0; aim[kk] = t1; anm[kk] = t2;
            }

            const int p       = 8 - 4 * g;          // group bit position
            const int lowmask = (1 << p) - 1;

            for (int t = wave; t < 16; t += 8) {    // wave-uniform tile loop
                const int c     = t * 16 + r;       // column index 0..255
                const int cpart = ((c >> p) << (p + 4)) | (c & lowmask);

                // ---- load B fragments from LDS (re & im) ----
                v2f bre[4], bim[4];
                #pragma unroll
                for (int kk = 0; kk < 4; ++kk) {
                    int m0 = 4 * kk + colb;
                    int a0 = cpart | (m0 << p);
                    int a1 = cpart | ((m0 + 1) << p);
                    v2f tr, ti;
                    tr[0] = S[cur + a0];        tr[1] = S[cur + a1];
                    ti[0] = S[cur + 4096 + a0]; ti[1] = S[cur + 4096 + a1];
                    bre[kk] = tr; bim[kk] = ti;
                }

                // ---- complex 16x16x16 matmul via 16 f32 WMMAs ----
                v8f accre = {};
                v8f accim = {};
                #pragma unroll
                for (int kk = 0; kk < 4; ++kk)
                    accre = __builtin_amdgcn_wmma_f32_16x16x4_f32(
                        false, are[kk], false, bre[kk], (short)0, accre, false, false);
                #pragma unroll
                for (int kk = 0; kk < 4; ++kk)
                    accre = __builtin_amdgcn_wmma_f32_16x16x4_f32(
                        false, anm[kk], false, bim[kk], (short)0, accre, false, false);
                #pragma unroll
                for (int kk = 0; kk < 4; ++kk)
                    accim = __builtin_amdgcn_wmma_f32_16x16x4_f32(
                        false, are[kk], false, bim[kk], (short)0, accim, false, false);
                #pragma unroll
                for (int kk = 0; kk < 4; ++kk)
                    accim = __builtin_amdgcn_wmma_f32_16x16x4_f32(
                        false, aim[kk], false, bre[kk], (short)0, accim, false, false);

                // ---- writeback; last group writes permuted into other buffer
                if (g < 2) {
                    #pragma unroll
                    for (int v = 0; v < 8; ++v) {
                        int ad = cpart | ((v + mb) << p);
                        S[cur + ad]        = accre[v];
                        S[cur + 4096 + ad] = accim[v];
                    }
                } else {
                    #pragma unroll
                    for (int v = 0; v < 8; ++v) {
                        int ad = cpart | ((v + mb) << p);
                        int dd = PI[ad];
                        S[nxt + dd]        = accre[v];
                        S[nxt + 4096 + dd] = accim[v];
                    }
                }
            }
            __syncthreads();
        }
    }

    // ---- readout: probs over 4 ancilla outcomes (i & 3 == tid & 3 per thread)
    const int fin = 8192;   // after 15 layers state is in buffer 1
    float pl = 0.f;
    #pragma unroll
    for (int k = 0; k < 16; ++k) {
        int i = tid + 256 * k;
        float re = S[fin + i], im = S[fin + 4096 + i];
        pl += re * re + im * im;
    }
    rbuf[tid] = pl;
    __syncthreads();
    if (tid < 4) {
        float s = 0.f;
        for (int k = tid; k < 256; k += 4) s += rbuf[k];
        rbuf[256 + tid] = s;
    }
    __syncthreads();
    if (tid < 4) {
        float tot = rbuf[256] + rbuf[257] + rbuf[258] + rbuf[259];
        out[b * 4 + tid] = rbuf[256 + tid] / tot;
    }
}

// ---------------------------------------------------------------------------
extern "C" void kernel_launch(void* const* d_in, const int* in_sizes, int n_in,
                              void* d_out, int out_size, void* d_ws, size_t ws_size,
                              hipStream_t stream) {
    const float* x = (const float*)d_in[0];   // (256,1,32,32) f32
    const float* w = (const float*)d_in[1];   // (540,) f32
    float* out = (float*)d_out;               // (256,4) f32
    float* ws  = (float*)d_ws;                // 45*768 floats = 135 KB

    qnn_prep<<<dim3(DEPTH * 3), dim3(256), 0, stream>>>(w, ws);
    qnn_main<<<dim3(256), dim3(256), 0, stream>>>(x, ws, out);
}